// GAT_47510928228567
// MI455X (gfx1250) — compile-verified
//
#include <hip/hip_runtime.h>
#include <math.h>

#define NNODES 50000
#define KNBR   16
#define FIN    140
#define HEADS  8
#define DHEAD  16
#define HID    128
#define NEG_SLOPE 0.2f

typedef __attribute__((ext_vector_type(2))) float v2f;
typedef __attribute__((ext_vector_type(8))) float v8f;

// out[M,128] = act(A[M,Kd] @ B[Kd,128] + bias), act: relu or identity.
// One 16-row tile per block; 8 waves, each wave owns one 16-col tile.
// FP32 WMMA 16x16x4: A fragment v2f (lanes 0-15: K0/K1, lanes 16-31: K2/K3),
// B fragment mirrored on the N axis, C/D = 8 VGPRs per ISA layout.
__global__ __launch_bounds__(256) void gemm_wmma_f32(
    const float* __restrict__ A, const float* __restrict__ B,
    const float* __restrict__ bias, float* __restrict__ out,
    int Kd, int relu)
{
  const int lane = threadIdx.x & 31;
  const int wid  = threadIdx.x >> 5;     // 0..7 -> column tile
  const int m0   = blockIdx.x << 4;      // 16 rows per block
  const int n0   = wid << 4;
  const int r16  = lane & 15;            // row (A) / col (B,C,D) within tile
  const int hi   = lane >> 4;            // 0 or 1
  const int koff = hi << 1;              // 0 or 2

  const float* Ap = A + (size_t)(m0 + r16) * Kd;
  const float* Bp = B + n0 + r16;

  v8f c = {};
  for (int k = 0; k < Kd; k += 4) {
    v2f a, b;
    a[0] = Ap[k + koff];
    a[1] = Ap[k + koff + 1];
    b[0] = Bp[(size_t)(k + koff) * HID];
    b[1] = Bp[(size_t)(k + koff + 1) * HID];
    // 8 args: (neg_a, A, neg_b, B, c_mod, C, reuse_a, reuse_b)
    c = __builtin_amdgcn_wmma_f32_16x16x4_f32(false, a, false, b,
                                              (short)0, c, false, false);
  }

  const float bv    = bias[n0 + r16];
  const int   mbase = m0 + (hi << 3);
#pragma unroll
  for (int r = 0; r < 8; ++r) {
    float v = c[r] + bv;
    if (relu) v = fmaxf(v, 0.0f);
    out[(size_t)(mbase + r) * HID + n0 + r16] = v;
  }
}

// One thread per (node, head): s_src/s_dst = <h[node,head,:], a_{src,dst}[head,:]>
__global__ __launch_bounds__(256) void gat_scores(
    const float* __restrict__ h, const float* __restrict__ asrc,
    const float* __restrict__ adst, float* __restrict__ ssrc,
    float* __restrict__ sdst)
{
  const int idx = blockIdx.x * 256 + threadIdx.x;   // node*HEADS + head
  if (idx >= NNODES * HEADS) return;
  const int head = idx & (HEADS - 1);
  const int node = idx >> 3;
  const float* hp = h + (size_t)node * HID + head * DHEAD;
  const float* as = asrc + head * DHEAD;
  const float* ad = adst + head * DHEAD;
  float s1 = 0.0f, s2 = 0.0f;
#pragma unroll
  for (int d = 0; d < DHEAD; ++d) {
    const float v = hp[d];
    s1 += v * as[d];
    s2 += v * ad[d];
  }
  ssrc[idx] = s1;
  sdst[idx] = s2;
}

__device__ __forceinline__ float elu1(float x) {
  return x > 0.0f ? x : expm1f(x);
}

// One wave per node. 4 lanes per head, each lane owns 4 contiguous channels.
// Per-lane redundant softmax over K=16 neighbor scores (L0/L2-resident).
__global__ __launch_bounds__(256) void gat_attn(
    const float* __restrict__ h, const float* __restrict__ ssrc,
    const float* __restrict__ sdst, const int* __restrict__ nb,
    float* __restrict__ out)
{
  const int gtid = blockIdx.x * 256 + threadIdx.x;
  const int node = gtid >> 5;
  if (node >= NNODES) return;
  const int lane = threadIdx.x & 31;
  const int head = lane >> 2;            // 8 heads over 32 lanes
  const int d0   = (lane & 3) << 2;      // 4 channels per lane

  int nbr[KNBR];
  const int4* nbv = (const int4*)(nb + (size_t)node * KNBR);
#pragma unroll
  for (int q = 0; q < KNBR / 4; ++q) {
    const int4 t = nbv[q];
    nbr[4 * q + 0] = t.x; nbr[4 * q + 1] = t.y;
    nbr[4 * q + 2] = t.z; nbr[4 * q + 3] = t.w;
  }

  const float s0 = ssrc[(size_t)node * HEADS + head];
  float e[KNBR];
  float emax = -3.0e38f;
#pragma unroll
  for (int k = 0; k < KNBR; ++k) {
    float s = s0 + sdst[(size_t)nbr[k] * HEADS + head];
    s = (s > 0.0f) ? s : NEG_SLOPE * s;   // leaky_relu(0.2)
    e[k] = s;
    emax = fmaxf(emax, s);
  }
  float denom = 0.0f;
#pragma unroll
  for (int k = 0; k < KNBR; ++k) {
    e[k] = __expf(e[k] - emax);
    denom += e[k];
  }
  const float inv = 1.0f / denom;

  float4 acc = make_float4(0.0f, 0.0f, 0.0f, 0.0f);
#pragma unroll
  for (int k = 0; k < KNBR; ++k) {
    const float4 hv =
        *(const float4*)(h + (size_t)nbr[k] * HID + head * DHEAD + d0);
    const float a = e[k] * inv;
    acc.x += a * hv.x; acc.y += a * hv.y;
    acc.z += a * hv.z; acc.w += a * hv.w;
  }
  acc.x = elu1(acc.x); acc.y = elu1(acc.y);
  acc.z = elu1(acc.z); acc.w = elu1(acc.w);
  *(float4*)(out + (size_t)node * HID + head * DHEAD + d0) = acc;
}

extern "C" void kernel_launch(void* const* d_in, const int* in_sizes, int n_in,
                              void* d_out, int out_size, void* d_ws, size_t ws_size,
                              hipStream_t stream) {
  (void)in_sizes; (void)n_in; (void)out_size; (void)ws_size;

  const float* feature = (const float*)d_in[0];   // [N, 140]
  const int*   nb_id   = (const int*)  d_in[1];   // [N, 16]
  const float* pca_w   = (const float*)d_in[2];   // [140, 128]
  const float* pca_b   = (const float*)d_in[3];   // [128]
  const float* Ws      = (const float*)d_in[4];   // [2, 128, 128]
  const float* bs      = (const float*)d_in[5];   // [2, 128]
  const float* a_src   = (const float*)d_in[6];   // [2, 8, 16]
  const float* a_dst   = (const float*)d_in[7];   // [2, 8, 16]
  float* out = (float*)d_out;                     // [N, 128]

  // Workspace: xB (N*HID) | h (N*HID) | s_src (N*H) | s_dst (N*H) = ~54.4 MB
  char*  ws  = (char*)d_ws;
  float* xB  = (float*)ws;
  float* hb  = (float*)(ws + (size_t)NNODES * HID * sizeof(float));
  float* ssr = (float*)(ws + (size_t)NNODES * HID * sizeof(float) * 2);
  float* sds = ssr + (size_t)NNODES * HEADS;

  const dim3 blk(256);
  const dim3 gGemm(NNODES / 16);                       // 3125 (exact)
  const dim3 gSc((NNODES * HEADS + 255) / 256);        // 1563
  const dim3 gAt((NNODES * 32 + 255) / 256);           // 6250 (wave per node)

  // x0 = relu(feature @ pca_w + pca_b) -> staged in d_out (overwritten later)
  gemm_wmma_f32<<<gGemm, blk, 0, stream>>>(feature, pca_w, pca_b, out, FIN, 1);

  // ---- layer 0 ----
  gemm_wmma_f32<<<gGemm, blk, 0, stream>>>(out, Ws, bs, hb, HID, 0);
  gat_scores<<<gSc, blk, 0, stream>>>(hb, a_src, a_dst, ssr, sds);
  gat_attn<<<gAt, blk, 0, stream>>>(hb, ssr, sds, nb_id, xB);

  // ---- layer 1 ----
  gemm_wmma_f32<<<gGemm, blk, 0, stream>>>(xB, Ws + HID * HID, bs + HID, hb, HID, 0);
  gat_scores<<<gSc, blk, 0, stream>>>(hb, a_src + HEADS * DHEAD,
                                      a_dst + HEADS * DHEAD, ssr, sds);
  gat_attn<<<gAt, blk, 0, stream>>>(hb, ssr, sds, nb_id, out);
}